// TwoLayerLSTMOptimizer_11493332484832
// MI455X (gfx1250) — compile-verified
//
#include <hip/hip_runtime.h>
#include <math.h>

// ---------------------------------------------------------------------------
// Problem collapse (D=1): u = [sign(bx),sign(bl),sign(bg)] has 27 states, and
// both LSTM recurrent terms use constant learned initial states. So
//   out[i] = (polyval * gamma^t) * tanh_v[trit_index] * (|bx|>0 ? |bx| : 1)
// Kernel 1 (1 wave): build 27-entry LUT; layer-1 gates via V_WMMA_F32_16X16X4_F32.
// Kernel 2 (streaming): 16 MB traffic, memory-bound (~0.7us @ 23.3 TB/s).
// ---------------------------------------------------------------------------

typedef __attribute__((ext_vector_type(2))) float v2f;
typedef __attribute__((ext_vector_type(4))) float f4;
typedef __attribute__((ext_vector_type(8))) float v8f;

__device__ __forceinline__ float sigm_(float x) { return 1.0f / (1.0f + expf(-x)); }

// u component for combo m (0..26): digits base-3, x=m/9, l=m/3, g=m, value d-1.
__device__ __forceinline__ float trit_(int m, int j) {
  int d = (j == 0) ? (m / 9) % 3 : (j == 1) ? (m / 3) % 3 : (m % 3);
  return (float)(d - 1);
}

__global__ __launch_bounds__(32)
void build_lut_kernel(const float* __restrict__ W_ih1, const float* __restrict__ W_hh1,
                      const float* __restrict__ b_ih1, const float* __restrict__ b_hh1,
                      const float* __restrict__ W_ih2, const float* __restrict__ W_hh2,
                      const float* __restrict__ b_ih2, const float* __restrict__ b_hh2,
                      const float* __restrict__ W_out, const float* __restrict__ b_out,
                      const float* __restrict__ h1_0, const float* __restrict__ c1_0,
                      const float* __restrict__ h2_0, const float* __restrict__ c2_0,
                      const float* __restrict__ poly, const int* __restrict__ t_ptr,
                      float* __restrict__ lut) {
  __shared__ float c1const[20];
  __shared__ float c2const[20];
  __shared__ float g1s[32][20];

  const int l  = threadIdx.x;   // 0..31, one wave
  const int r  = l & 15;
  const int hi = l >> 4;

  // Batch-independent gate constants: b_ih + b_hh + h0 @ W_hh^T
  if (l < 20) {
    float a = b_ih1[l] + b_hh1[l];
    float b = b_ih2[l] + b_hh2[l];
#pragma unroll
    for (int k = 0; k < 5; ++k) {
      a += W_hh1[l * 5 + k] * h1_0[k];
      b += W_hh2[l * 5 + k] * h2_0[k];
    }
    c1const[l] = a;
    c2const[l] = b;
  }
  __syncthreads();

  // ----- A tiles: trit combos, 16x4 f32 layout (lanes0-15: K0,K1; lanes16-31: K2,K3=0)
  v2f a0, a1, b0, b1;
  const int m0 = r, m1 = 16 + r;
  if (hi == 0) {
    a0.x = trit_(m0, 0); a0.y = trit_(m0, 1);
    a1.x = trit_(m1, 0); a1.y = trit_(m1, 1);
  } else {
    a0.x = trit_(m0, 2); a0.y = 0.0f;
    a1.x = trit_(m1, 2); a1.y = 0.0f;
  }
  // ----- B tiles: W_ih1^T (K=3 padded to 4), cols = gate index; B[k][n]=W_ih1[n][k]
  const int g16 = 16 + r;
  if (hi == 0) {
    b0.x = W_ih1[r * 3 + 0];
    b0.y = W_ih1[r * 3 + 1];
    b1.x = (g16 < 20) ? W_ih1[g16 * 3 + 0] : 0.0f;
    b1.y = (g16 < 20) ? W_ih1[g16 * 3 + 1] : 0.0f;
  } else {
    b0.x = W_ih1[r * 3 + 2];
    b0.y = 0.0f;
    b1.x = (g16 < 20) ? W_ih1[g16 * 3 + 2] : 0.0f;
    b1.y = 0.0f;
  }

  v8f z = {};
  // gates1[32 combos][20 gates] = U(32x4) @ Wih1^T(4x32cols) as four 16x16 WMMA tiles
  v8f c00 = __builtin_amdgcn_wmma_f32_16x16x4_f32(false, a0, false, b0, (short)0, z, false, false);
  v8f c01 = __builtin_amdgcn_wmma_f32_16x16x4_f32(false, a0, false, b1, (short)0, z, false, false);
  v8f c10 = __builtin_amdgcn_wmma_f32_16x16x4_f32(false, a1, false, b0, (short)0, z, false, false);
  v8f c11 = __builtin_amdgcn_wmma_f32_16x16x4_f32(false, a1, false, b1, (short)0, z, false, false);

  // Scatter C tiles to LDS (+const): lane L, vgpr v -> (M = v + 8*(L>=16), N = L%16)
  const int colA = r;
  const int colB = 16 + r;
#pragma unroll
  for (int v = 0; v < 8; ++v) {
    const int row0 = v + 8 * hi;        // A0 tile rows (combos 0..15)
    const int row1 = 16 + v + 8 * hi;   // A1 tile rows (combos 16..31)
    g1s[row0][colA] = c00[v] + c1const[colA];
    g1s[row1][colA] = c10[v] + c1const[colA];
    if (colB < 20) {
      g1s[row0][colB] = c01[v] + c1const[colB];
      g1s[row1][colB] = c11[v] + c1const[colB];
    }
  }
  __syncthreads();

  // Per-lane (27 active): activations + layer 2 + head, fold poly*gamma^t into LUT.
  if (l < 27) {
    float h1v[5];
#pragma unroll
    for (int k = 0; k < 5; ++k) {
      const float ig = sigm_(g1s[l][k]);
      const float fg = sigm_(g1s[l][5 + k]);
      const float gg = tanhf(g1s[l][10 + k]);
      const float og = sigm_(g1s[l][15 + k]);
      const float cn = fg * c1_0[k] + ig * gg;
      h1v[k] = og * tanhf(cn);
    }
    float h2v[5];
#pragma unroll
    for (int k = 0; k < 5; ++k) {
      float gi = c2const[k], gf = c2const[5 + k], gg = c2const[10 + k], go = c2const[15 + k];
#pragma unroll
      for (int j = 0; j < 5; ++j) {
        gi += W_ih2[(k) * 5 + j] * h1v[j];
        gf += W_ih2[(5 + k) * 5 + j] * h1v[j];
        gg += W_ih2[(10 + k) * 5 + j] * h1v[j];
        go += W_ih2[(15 + k) * 5 + j] * h1v[j];
      }
      const float cn = sigm_(gf) * c2_0[k] + sigm_(gi) * tanhf(gg);
      h2v[k] = sigm_(go) * tanhf(cn);
    }
    float v = b_out[0];
#pragma unroll
    for (int k = 0; k < 5; ++k) v += W_out[k] * h2v[k];
    const float tv = tanhf(v);

    const int t = t_ptr[0];
    float pv = 0.0f, tp = 1.0f;   // t^0 == 1 even for t==0 (matches Python 0**0)
#pragma unroll
    for (int j = 0; j < 4; ++j) {
      pv += logf(1.0f + expf(poly[j])) * tp;  // softplus
      tp *= (float)t;
    }
    const float F = pv * powf(0.99f, (float)t);
    lut[l] = F * tv;
  }
}

// ---------------------------------------------------------------------------

__device__ __forceinline__ float apply1_(float x, float lv, float gv,
                                         const float* __restrict__ slut) {
  const int idx = (x > 0.0f ? 18 : (x < 0.0f ? 0 : 9)) +
                  (lv > 0.0f ? 6 : (lv < 0.0f ? 0 : 3)) +
                  (gv > 0.0f ? 2 : (gv < 0.0f ? 0 : 1));
  const float ax = fabsf(x);
  const float s = (ax > 0.0f) ? ax : 1.0f;
  return slut[idx] * s;   // ds_load gather from 27-word LDS (one word per bank)
}

__global__ __launch_bounds__(256)
void apply_kernel(const float* __restrict__ bx, const float* __restrict__ bl,
                  const float* __restrict__ bg, const float* __restrict__ lut,
                  float* __restrict__ out, int n) {
  __shared__ float slut[27];
  if (threadIdx.x < 27) slut[threadIdx.x] = lut[threadIdx.x];
  __syncthreads();

  const int n4 = n >> 2;
  const int i4 = blockIdx.x * blockDim.x + threadIdx.x;
  if (i4 < n4) {
    // Streamed once: nontemporal 128-bit accesses, fully coalesced.
    const f4 x = __builtin_nontemporal_load((const f4*)bx + i4);
    const f4 l = __builtin_nontemporal_load((const f4*)bl + i4);
    const f4 g = __builtin_nontemporal_load((const f4*)bg + i4);
    f4 o;
    o.x = apply1_(x.x, l.x, g.x, slut);
    o.y = apply1_(x.y, l.y, g.y, slut);
    o.z = apply1_(x.z, l.z, g.z, slut);
    o.w = apply1_(x.w, l.w, g.w, slut);
    __builtin_nontemporal_store(o, (f4*)out + i4);
  }
  // Scalar tail (B is a multiple of 4 in practice; kept for generality).
  const int rem = n & 3;
  if (blockIdx.x == 0 && (int)threadIdx.x < rem) {
    const int i = (n & ~3) + threadIdx.x;
    out[i] = apply1_(bx[i], bl[i], bg[i], slut);
  }
}

// ---------------------------------------------------------------------------

extern "C" void kernel_launch(void* const* d_in, const int* in_sizes, int n_in,
                              void* d_out, int out_size, void* d_ws, size_t ws_size,
                              hipStream_t stream) {
  const float* bx    = (const float*)d_in[0];
  const float* bl    = (const float*)d_in[1];
  const float* bg    = (const float*)d_in[2];
  const float* W_ih1 = (const float*)d_in[3];
  const float* W_hh1 = (const float*)d_in[4];
  const float* b_ih1 = (const float*)d_in[5];
  const float* b_hh1 = (const float*)d_in[6];
  const float* W_ih2 = (const float*)d_in[7];
  const float* W_hh2 = (const float*)d_in[8];
  const float* b_ih2 = (const float*)d_in[9];
  const float* b_hh2 = (const float*)d_in[10];
  const float* W_out = (const float*)d_in[11];
  const float* b_out = (const float*)d_in[12];
  const float* h1_0  = (const float*)d_in[13];
  const float* c1_0  = (const float*)d_in[14];
  const float* h2_0  = (const float*)d_in[15];
  const float* c2_0  = (const float*)d_in[16];
  const float* poly  = (const float*)d_in[17];
  const int*   t_ptr = (const int*)d_in[18];

  float* lut = (float*)d_ws;           // 27 floats of scratch
  float* out = (float*)d_out;
  const int n = in_sizes[0];           // B (D == 1)

  build_lut_kernel<<<1, 32, 0, stream>>>(W_ih1, W_hh1, b_ih1, b_hh1,
                                         W_ih2, W_hh2, b_ih2, b_hh2,
                                         W_out, b_out, h1_0, c1_0, h2_0, c2_0,
                                         poly, t_ptr, lut);

  int n4 = n >> 2;
  int blocks = (n4 + 255) / 256;
  if (blocks < 1) blocks = 1;
  apply_kernel<<<blocks, 256, 0, stream>>>(bx, bl, bg, lut, out, n);
}